// IDCT_1133871366827
// MI455X (gfx1250) — compile-verified
//
#include <hip/hip_runtime.h>
#include <math.h>

// IDCT (DCT-III) over rows of a 4096x4096 f32 matrix, recast as
//   Y = X * W,  W[n][k] = c_n * cos(pi * n * (2k+1) / (2N))
// executed with full-precision f32 WMMA (v_wmma_f32_16x16x4_f32).
//
// Round 3: software-pipelined staging — global_load_b128 for tile k+1 issue
// BEFORE the 64-WMMA compute phase of tile k, so HBM/L2 latency is hidden
// behind the matrix pipe instead of being exposed at every barrier.

typedef __attribute__((ext_vector_type(2))) float v2f;
typedef __attribute__((ext_vector_type(4))) float v4f;
typedef __attribute__((ext_vector_type(8))) float v8f;

#define DCT_N 4096
#define BM 128
#define BN 128
#define BK 32
#define LDA 36     // padded LDS row stride (floats) for A tile (32 cols + 4 pad)
#define LDBP 260   // padded LDS stride (floats) per K-pair row of B (256 + 4 pad)

// ---------------------------------------------------------------------------
// Build the DCT-III coefficient matrix W (row-major [n][k], 4096x4096 f32).
// Exact integer range reduction: cos(pi * p / (2N)) has period p mod 4N.
// ---------------------------------------------------------------------------
__global__ __launch_bounds__(256) void idct_build_w(float* __restrict__ W) {
  int idx = blockIdx.x * 256 + threadIdx.x;       // 0 .. N*N-1
  int n = idx >> 12;                              // idx / 4096
  int k = idx & (DCT_N - 1);                      // idx % 4096
  unsigned prod = (unsigned)n * (unsigned)(2 * k + 1);
  unsigned r = prod & (4u * DCT_N - 1u);          // mod 4N (power of two)
  const float step = 3.14159265358979323846f / (2.0f * (float)DCT_N);
  float c = cosf((float)r * step);                // arg in [0, 2*pi)
  W[idx] = (n == 0 ? 1.0f : 2.0f) * c;
}

// ---------------------------------------------------------------------------
// Tiled f32 WMMA GEMM: Y[M,N] = X[M,K] * W[K,N], M=N=K=4096.
// Block: 128x128 tile, 256 threads = 8 waves in a 4(M) x 2(N) grid;
// each wave computes a 32x64 patch = 2x4 tiles of 16x16 via WMMA 16x16x4.
// ---------------------------------------------------------------------------
__global__ __launch_bounds__(256) void idct_gemm(const float* __restrict__ X,
                                                 const float* __restrict__ W,
                                                 float* __restrict__ Y) {
  __shared__ float As[BM * LDA];          // 128*36*4  = 18432 B
  __shared__ float Bs[(BK / 2) * LDBP];   // 16*260*4  = 16640 B (K-pair layout)

  const int tid  = threadIdx.x;
  const int lane = tid & 31;       // wave32
  const int wave = tid >> 5;       // 0..7
  const int half = lane >> 4;      // 0 or 1 (lane group per WMMA layout)
  const int lid  = lane & 15;

  const int tileM = blockIdx.y * BM;
  const int tileN = blockIdx.x * BN;

  const int wm = (wave & 3) * 32;  // wave row offset inside block tile
  const int wn = (wave >> 2) * 64; // wave col offset inside block tile

  // Loop-invariant staging geometry.
  // A: 1024 float4 per tile, 4 per thread (8 float4 per 32-float row).
  int aR[4], aC[4];
#pragma unroll
  for (int i = 0; i < 4; ++i) {
    int f = tid + i * 256;
    aR[i] = f >> 3;
    aC[i] = (f & 7) << 2;
  }
  // B: 512 tasks (K-pair row, 4-col group), 2 per thread.
  int bPR[2], bC[2];
#pragma unroll
  for (int i = 0; i < 2; ++i) {
    int task = tid + i * 256;
    bPR[i] = task >> 5;
    bC[i]  = (task & 31) << 2;
  }

  v8f acc[2][4];
#pragma unroll
  for (int i = 0; i < 2; ++i)
#pragma unroll
    for (int j = 0; j < 4; ++j)
      acc[i][j] = (v8f){0.f, 0.f, 0.f, 0.f, 0.f, 0.f, 0.f, 0.f};

  // Per-wave fragment base pointers into LDS.
  // A frag (16x4): lane(0-15) K0,K1 ; lane(16-31) K2,K3 ; row = m = lid
  const float* Abase = &As[(wm + lid) * LDA + 2 * half];
  // B frag (4x16): pair row p holds interleaved {B[2p][n], B[2p+1][n]} at 2n.
  const float* Bbase = &Bs[half * LDBP + 2 * (wn + lid)];

  // ---- Prologue: stage tile k0 = 0 into LDS ----
#pragma unroll
  for (int i = 0; i < 4; ++i) {
    v4f v = *(const v4f*)(X + (size_t)(tileM + aR[i]) * DCT_N + aC[i]);
    *(v4f*)(&As[aR[i] * LDA + aC[i]]) = v;
  }
#pragma unroll
  for (int i = 0; i < 2; ++i) {
    const float* g = W + (size_t)(2 * bPR[i]) * DCT_N + (tileN + bC[i]);
    v4f lo = *(const v4f*)(g);
    v4f hi = *(const v4f*)(g + DCT_N);
    float* s = &Bs[bPR[i] * LDBP + 2 * bC[i]];
    *(v4f*)(s + 0) = (v4f){lo.x, hi.x, lo.y, hi.y};
    *(v4f*)(s + 4) = (v4f){lo.z, hi.z, lo.w, hi.w};
  }
  __syncthreads();

  for (int k0 = 0; k0 < DCT_N; k0 += BK) {
    const bool has_next = (k0 + BK) < DCT_N;

    // ---- Prefetch next tile into registers (issues before the WMMAs) ----
    v4f pa[4], plo[2], phi[2];
    if (has_next) {
      const int kn = k0 + BK;
#pragma unroll
      for (int i = 0; i < 4; ++i)
        pa[i] = *(const v4f*)(X + (size_t)(tileM + aR[i]) * DCT_N + (kn + aC[i]));
#pragma unroll
      for (int i = 0; i < 2; ++i) {
        const float* g = W + (size_t)(kn + 2 * bPR[i]) * DCT_N + (tileN + bC[i]);
        plo[i] = *(const v4f*)(g);
        phi[i] = *(const v4f*)(g + DCT_N);
      }
    }

    // ---- Compute: 8 K-steps of 4, 8 WMMAs per step per wave ----
#pragma unroll
    for (int kk = 0; kk < BK; kk += 4) {
      v2f a[2], b[4];
#pragma unroll
      for (int i = 0; i < 2; ++i)
        a[i] = *(const v2f*)(Abase + 16 * i * LDA + kk);         // ds_load_b64
#pragma unroll
      for (int j = 0; j < 4; ++j)
        b[j] = *(const v2f*)(Bbase + (kk >> 1) * LDBP + 32 * j); // ds_load_b64
#pragma unroll
      for (int i = 0; i < 2; ++i)
#pragma unroll
        for (int j = 0; j < 4; ++j)
          acc[i][j] = __builtin_amdgcn_wmma_f32_16x16x4_f32(
              /*neg_a=*/false, a[i], /*neg_b=*/false, b[j],
              /*c_mod=*/(short)0, acc[i][j],
              /*reuse_a=*/false, /*reuse_b=*/false);
    }
    __syncthreads();

    // ---- Commit prefetched tile to LDS ----
    if (has_next) {
#pragma unroll
      for (int i = 0; i < 4; ++i)
        *(v4f*)(&As[aR[i] * LDA + aC[i]]) = pa[i];
#pragma unroll
      for (int i = 0; i < 2; ++i) {
        float* s = &Bs[bPR[i] * LDBP + 2 * bC[i]];
        *(v4f*)(s + 0) = (v4f){plo[i].x, phi[i].x, plo[i].y, phi[i].y};
        *(v4f*)(s + 4) = (v4f){plo[i].z, phi[i].z, plo[i].w, phi[i].w};
      }
    }
    __syncthreads();
  }

  // ---- Store: C/D layout -> VGPR v holds row (v + 8*half), col = lid ----
#pragma unroll
  for (int i = 0; i < 2; ++i) {
    int rowBase = tileM + wm + 16 * i + 8 * half;
#pragma unroll
    for (int j = 0; j < 4; ++j) {
      int col = tileN + wn + 16 * j + lid;
#pragma unroll
      for (int v = 0; v < 8; ++v)
        Y[(size_t)(rowBase + v) * DCT_N + col] = acc[i][j][v];
    }
  }
}

extern "C" void kernel_launch(void* const* d_in, const int* in_sizes, int n_in,
                              void* d_out, int out_size, void* d_ws, size_t ws_size,
                              hipStream_t stream) {
  (void)in_sizes; (void)n_in; (void)out_size; (void)ws_size;
  const float* X = (const float*)d_in[0];
  float* Y = (float*)d_out;
  float* W = (float*)d_ws;  // needs 4096*4096*4 = 64 MiB of scratch

  // 1) materialize the cosine matrix (deterministic, rebuilt every call)
  idct_build_w<<<(DCT_N * DCT_N) / 256, 256, 0, stream>>>(W);

  // 2) Y = X * W via f32 WMMA GEMM
  dim3 grid(DCT_N / BN, DCT_N / BM);
  idct_gemm<<<grid, 256, 0, stream>>>(X, W, Y);
}